// SwarmNet_13932873908789
// MI455X (gfx1250) — compile-verified
//
#include <hip/hip_runtime.h>
#include <hip/hip_bf16.h>

// Problem constants (from reference)
#define D_IN   4
#define H_DIM  32
#define N_SAMP 128
#define T_IN   100
#define TP     94              // after three VALID k=3 convs: 100-6
#define M_ROWS (TP * N_SAMP)   // 12032 rows for all (·,32) GEMMs
#define LPAD   33              // LDS row stride (floats) to dodge bank conflicts

typedef __attribute__((ext_vector_type(2))) float v2f;
typedef __attribute__((ext_vector_type(8))) float v8f;

#if defined(__has_builtin)
#if __has_builtin(__builtin_amdgcn_global_load_async_to_lds_b32) && \
    __has_builtin(__builtin_amdgcn_s_wait_asynccnt)
#define USE_ASYNC_LDS 1
#endif
#endif
#ifndef USE_ASYNC_LDS
#define USE_ASYNC_LDS 0
#endif

#if USE_ASYNC_LDS
typedef __attribute__((address_space(1))) int async_gint;  // global int
typedef __attribute__((address_space(3))) int async_lint;  // LDS int
#endif

// ---------------------------------------------------------------------------
// Kernel 1: conv stack, one block per sample n, all intermediates in LDS.
// Writes h in layout h[(t*N + n)*32 + c] (row-major (M=12032, K=32)).
// ---------------------------------------------------------------------------
__global__ void conv_stack_kernel(const float* __restrict__ x,
                                  const float* __restrict__ w1, const float* __restrict__ b1,
                                  const float* __restrict__ w2, const float* __restrict__ b2,
                                  const float* __restrict__ w3, const float* __restrict__ b3,
                                  float* __restrict__ hout) {
    const int n   = blockIdx.x;      // 0..127
    const int tid = threadIdx.x;     // 0..127

    __shared__ float xs[T_IN * D_IN];      // 400 f
    __shared__ float h1[16 * 98];          // 1568 f
    __shared__ float h2[32 * 96];          // 3072 f  (~20 KB of 320 KB)

    for (int i = tid; i < T_IN * D_IN; i += 128) xs[i] = x[n * T_IN * D_IN + i];
    __syncthreads();

    for (int p = tid; p < 16 * 98; p += 128) {          // conv1: 4->16
        const int oc = p / 98, t = p % 98;
        float acc = b1[oc];
#pragma unroll
        for (int k = 0; k < 3; ++k)
#pragma unroll
            for (int ic = 0; ic < 4; ++ic)
                acc += xs[(t + k) * 4 + ic] * w1[oc * 12 + ic * 3 + k];
        h1[oc * 98 + t] = fmaxf(acc, 0.f);
    }
    __syncthreads();

    for (int p = tid; p < 32 * 96; p += 128) {          // conv2: 16->32
        const int oc = p / 96, t = p % 96;
        float acc = b2[oc];
        for (int ic = 0; ic < 16; ++ic)
#pragma unroll
            for (int k = 0; k < 3; ++k)
                acc += h1[ic * 98 + t + k] * w2[oc * 48 + ic * 3 + k];
        h2[oc * 96 + t] = fmaxf(acc, 0.f);
    }
    __syncthreads();

    for (int p = tid; p < 32 * TP; p += 128) {          // conv3: 32->32
        const int oc = p / TP, t = p % TP;
        float acc = b3[oc];
        for (int ic = 0; ic < 32; ++ic)
#pragma unroll
            for (int k = 0; k < 3; ++k)
                acc += h2[ic * 96 + t + k] * w3[oc * 96 + ic * 3 + k];
        hout[(t * N_SAMP + n) * H_DIM + oc] = fmaxf(acc, 0.f);
    }
}

// ---------------------------------------------------------------------------
// WMMA tile helper: one wave computes a 16x16 f32 tile via 8x 16x16x4 WMMA.
// A layout (ISA 7.12.2): lanes 0-15 K=k0..k0+1, lanes 16-31 K=k0+2..k0+3.
// C/D: vgpr r, lane l -> M = r + 8*(l>=16), N = l&15.
// ---------------------------------------------------------------------------
__device__ __forceinline__ v8f wmma_k32(const float* Arow /* per-lane row ptr */,
                                        const float* W /* (32,32) row-major */,
                                        int n0, int hi, int ml, v8f acc) {
#pragma unroll
    for (int k0 = 0; k0 < 32; k0 += 4) {
        v2f a, b;
        a.x = Arow[k0 + 2 * hi + 0];
        a.y = Arow[k0 + 2 * hi + 1];
        b.x = W[(k0 + 2 * hi + 0) * H_DIM + n0 + ml];
        b.y = W[(k0 + 2 * hi + 1) * H_DIM + n0 + ml];
        acc = __builtin_amdgcn_wmma_f32_16x16x4_f32(false, a, false, b,
                                                    (short)0, acc, false, false);
    }
    return acc;
}

// ---------------------------------------------------------------------------
// Kernel 2: ej = h@We[:32], ei = h@We[32:], fused.  grid (188,4) x 128 thr.
// ---------------------------------------------------------------------------
__global__ void eje_gemm_kernel(const float* __restrict__ h,
                                const float* __restrict__ We,
                                float* __restrict__ ej, float* __restrict__ ei) {
    const int wv   = threadIdx.x >> 5;
    const int lane = threadIdx.x & 31;
    const int hi = lane >> 4, ml = lane & 15;
    const int m0 = (blockIdx.x * 4 + wv) * 16;
    const int sel = blockIdx.y >> 1;            // 0 -> Wj/ej, 1 -> Wi/ei
    const int n0  = (blockIdx.y & 1) * 16;

    const float* Arow = h + (size_t)(m0 + ml) * H_DIM;
    __builtin_prefetch(Arow, 0, 0);             // global_prefetch_b8
    const float* W = We + sel * 32 * H_DIM;
    float* Dout = sel ? ei : ej;

    v8f acc = {};
    acc = wmma_k32(Arow, W, n0, hi, ml, acc);

    const int col = n0 + ml;
#pragma unroll
    for (int r = 0; r < 8; ++r)
        Dout[(size_t)(m0 + r + hi * 8) * H_DIM + col] = acc[r];
}

// ---------------------------------------------------------------------------
// Kernel 3: per timestep t —
//   aggp[i,h] = sum_j relu(ej[j,h] + ei[i,h] + be[h]) - diag   (in LDS)
//   agg[t]    = relu(aggp @ Wa + ba)                            (WMMA)
// ej/ei slabs staged global->LDS (async path when available), bank-padded.
// ---------------------------------------------------------------------------
__global__ void pairwise_agg_kernel(const float* __restrict__ ej,
                                    const float* __restrict__ ei,
                                    const float* __restrict__ be,
                                    const float* __restrict__ Wa,
                                    const float* __restrict__ ba,
                                    float* __restrict__ agg) {
    const int t   = blockIdx.x;         // 0..93
    const int tid = threadIdx.x;        // 0..255

    __shared__ float sj[N_SAMP * LPAD];
    __shared__ float si[N_SAMP * LPAD];
    __shared__ float sa[N_SAMP * LPAD];

    const float* ejt = ej + (size_t)t * N_SAMP * H_DIM;
    const float* eit = ei + (size_t)t * N_SAMP * H_DIM;

#if USE_ASYNC_LDS
    {
        for (int idx = tid; idx < N_SAMP * H_DIM; idx += 256) {
            const int r = idx >> 5, c = idx & 31;
            __builtin_amdgcn_global_load_async_to_lds_b32(
                (async_gint*)(ejt + idx), (async_lint*)(sj + r * LPAD + c), 0, 0);
            __builtin_amdgcn_global_load_async_to_lds_b32(
                (async_gint*)(eit + idx), (async_lint*)(si + r * LPAD + c), 0, 0);
        }
        __builtin_amdgcn_s_wait_asynccnt(0);
    }
#else
    for (int idx = tid; idx < N_SAMP * H_DIM; idx += 256) {
        const int r = idx >> 5, c = idx & 31;
        sj[r * LPAD + c] = ejt[idx];
        si[r * LPAD + c] = eit[idx];
    }
#endif
    __syncthreads();

    // ---- O(N^2) relu-sum, 16 i-values register-blocked per thread ----
    {
        const int hh = tid & 31;        // feature
        const int i0 = tid >> 5;        // 0..7; thread owns i = i0 + 8k
        const float bev = be[hh];
        float c[16], acc[16];
#pragma unroll
        for (int k = 0; k < 16; ++k) {
            c[k]   = si[(i0 + 8 * k) * LPAD + hh] + bev;
            acc[k] = 0.f;
        }
        for (int j = 0; j < N_SAMP; ++j) {
            const float v = sj[j * LPAD + hh];   // 1 ds_load feeds 16 relu-adds
#pragma unroll
            for (int k = 0; k < 16; ++k)
                acc[k] += fmaxf(v + c[k], 0.f);
        }
#pragma unroll
        for (int k = 0; k < 16; ++k) {
            const int i = i0 + 8 * k;
            sa[i * LPAD + hh] = acc[k] - fmaxf(sj[i * LPAD + hh] + c[k], 0.f);
        }
    }
    __syncthreads();

    // ---- agg = relu(sa @ Wa + ba) via WMMA, A sourced from padded LDS ----
    {
        const int wv = tid >> 5, lane = tid & 31;
        const int hi = lane >> 4, ml = lane & 15;
        for (int tile = wv; tile < 16; tile += 8) {
            const int m0 = (tile & 7) * 16;
            const int n0 = (tile >> 3) * 16;
            const float* Arow = &sa[(m0 + ml) * LPAD];   // LDS, stride LPAD dodges conflicts
            v8f acc = {};
#pragma unroll
            for (int k0 = 0; k0 < 32; k0 += 4) {
                v2f a, b;
                a.x = Arow[k0 + 2 * hi + 0];
                a.y = Arow[k0 + 2 * hi + 1];
                b.x = Wa[(k0 + 2 * hi + 0) * H_DIM + n0 + ml];
                b.y = Wa[(k0 + 2 * hi + 1) * H_DIM + n0 + ml];
                acc = __builtin_amdgcn_wmma_f32_16x16x4_f32(false, a, false, b,
                                                            (short)0, acc, false, false);
            }
            const int col = n0 + ml;
            const float bv = ba[col];
#pragma unroll
            for (int r = 0; r < 8; ++r) {
                const int row = m0 + r + hi * 8;
                agg[((size_t)t * N_SAMP + row) * H_DIM + col] = fmaxf(acc[r] + bv, 0.f);
            }
        }
    }
}

// ---------------------------------------------------------------------------
// Kernel 4: upd = relu(h@Wu[:32] + agg@Wu[32:] + bu)  (WMMA, tile -> LDS)
//           dec = upd @ Wd + bd ; out = x[:,6:6+TP] + dec^T
// Block: 256 thr = 8 waves; 4 m-subtiles x 2 col-halves of a 64x32 upd tile.
// upd never touches global memory.
// ---------------------------------------------------------------------------
__global__ void upd_dec_kernel(const float* __restrict__ h,
                               const float* __restrict__ agg,
                               const float* __restrict__ Wu,
                               const float* __restrict__ bu,
                               const float* __restrict__ Wd,
                               const float* __restrict__ bd,
                               const float* __restrict__ x,
                               float* __restrict__ out) {
    const int tid = threadIdx.x;          // 0..255
    __shared__ float su[64 * LPAD];       // 64-row upd block tile, padded

    {
        const int wv = tid >> 5, lane = tid & 31;
        const int hi = lane >> 4, ml = lane & 15;
        const int mt = wv & 3;                      // m-subtile 0..3
        const int n0 = (wv >> 2) * 16;              // col half
        const int m0 = blockIdx.x * 64 + mt * 16;   // global row base

        const float* Arow  = h   + (size_t)(m0 + ml) * H_DIM;
        const float* A2row = agg + (size_t)(m0 + ml) * H_DIM;
        __builtin_prefetch(Arow, 0, 0);

        v8f acc = {};
        acc = wmma_k32(Arow,  Wu,               n0, hi, ml, acc);  // h   @ Wu[:32]
        acc = wmma_k32(A2row, Wu + 32 * H_DIM,  n0, hi, ml, acc);  // agg @ Wu[32:]

        const int col = n0 + ml;
        const float bv = bu[col];
#pragma unroll
        for (int r = 0; r < 8; ++r)
            su[(mt * 16 + r + hi * 8) * LPAD + col] = fmaxf(acc[r] + bv, 0.f);
    }
    __syncthreads();

    // dec + residual: 256 threads = 64 rows x 4 output features
    {
        const int d  = tid & 3;
        const int lr = tid >> 2;                    // 0..63
        const int m  = blockIdx.x * 64 + lr;        // m = t*128 + n
        const int t  = m / N_SAMP;
        const int n  = m % N_SAMP;
        float acc = bd[d];
#pragma unroll 8
        for (int c = 0; c < H_DIM; ++c) acc += su[lr * LPAD + c] * Wd[c * D_IN + d];
        out[((size_t)n * TP + t) * D_IN + d] =
            x[((size_t)n * T_IN + 6 + t) * D_IN + d] + acc;
    }
}

// ---------------------------------------------------------------------------
extern "C" void kernel_launch(void* const* d_in, const int* in_sizes, int n_in,
                              void* d_out, int out_size, void* d_ws, size_t ws_size,
                              hipStream_t stream) {
    const float* x  = (const float*)d_in[0];
    const float* w1 = (const float*)d_in[1];
    const float* b1 = (const float*)d_in[2];
    const float* w2 = (const float*)d_in[3];
    const float* b2 = (const float*)d_in[4];
    const float* w3 = (const float*)d_in[5];
    const float* b3 = (const float*)d_in[6];
    const float* We = (const float*)d_in[7];   // (64,32): rows 0-31 = Wj, 32-63 = Wi
    const float* be = (const float*)d_in[8];
    const float* Wa = (const float*)d_in[9];
    const float* ba = (const float*)d_in[10];
    const float* Wu = (const float*)d_in[11];  // (64,32): rows 0-31 for h, 32-63 for agg
    const float* bu = (const float*)d_in[12];
    const float* Wd = (const float*)d_in[13];
    const float* bd = (const float*)d_in[14];
    float* out = (float*)d_out;

    // Workspace: 4 slabs of M_ROWS*32 floats (~1.54 MB each)
    const size_t S = (size_t)M_ROWS * H_DIM;
    float* ws  = (float*)d_ws;
    float* h   = ws;
    float* ej  = ws + S;
    float* ei  = ws + 2 * S;
    float* agg = ws + 3 * S;

    // 1) conv stack -> h
    conv_stack_kernel<<<N_SAMP, 128, 0, stream>>>(x, w1, b1, w2, b2, w3, b3, h);

    // 2) ej / ei (fused, 1504 wave-tiles per output)
    eje_gemm_kernel<<<dim3(M_ROWS / 64, 4), 128, 0, stream>>>(h, We, ej, ei);

    // 3) pairwise relu-sum + Wa GEMM (WMMA from LDS) -> agg
    pairwise_agg_kernel<<<TP, 256, 0, stream>>>(ej, ei, be, Wa, ba, agg);

    // 4) Wu GEMM (WMMA) + dec + residual -> out
    upd_dec_kernel<<<M_ROWS / 64, 256, 0, stream>>>(h, agg, Wu, bu, Wd, bd, x, out);
}